// MemoryAttention_33483565040103
// MI455X (gfx1250) — compile-verified
//
#include <hip/hip_runtime.h>
#include <stdint.h>

typedef __attribute__((ext_vector_type(2))) float v2f;
typedef __attribute__((ext_vector_type(8))) float v8f;

#define NCTR   32768
#define DKEY   64
#define DVAL   128
#define DMODEL 256
#define ROWS_TOTAL 8192     // B*T = 4*2048
#define TOPK   32
#define WAVES  4
#define TILE_N 64           // centers per staged tile (4 x 16 sub-tiles)
#define NITER  (NCTR / TILE_N)
#define MU_STRIDE 68        // padded dword stride: 16B-aligned rows, conflict-free b64 reads

// --- async global->LDS copies (ASYNCcnt-tracked DMA, CDNA5) -----------------
// Low 32 bits of a flat pointer to a __shared__ object == LDS byte offset
// (ISA 10.2: LDS aperture uses addr[31:0] truncated).
__device__ __forceinline__ void async_g2l_b128(const float* g, float* l) {
    unsigned loff = (unsigned)(uintptr_t)(void*)l;
    unsigned long long ga = (unsigned long long)(uintptr_t)(const void*)g;
    asm volatile("global_load_async_to_lds_b128 %0, %1, off"
                 :: "v"(loff), "v"(ga) : "memory");
}
__device__ __forceinline__ void async_g2l_b32(const float* g, float* l) {
    unsigned loff = (unsigned)(uintptr_t)(void*)l;
    unsigned long long ga = (unsigned long long)(uintptr_t)(const void*)g;
    asm volatile("global_load_async_to_lds_b32 %0, %1, off"
                 :: "v"(loff), "v"(ga) : "memory");
}

// ---------------------------------------------------------------------------
// Kernel 0: per-center bias = -0.5*||mu_n||^2 / sigma^2   (q^2 term dropped:
// softmax weights and top-k order are invariant to per-row constants)
// ---------------------------------------------------------------------------
__global__ __launch_bounds__(128)
void memattn_bias_kernel(const float* __restrict__ mu,
                         const float* __restrict__ sigma,
                         float* __restrict__ bias) {
    int n = blockIdx.x * blockDim.x + threadIdx.x;
    if (n >= NCTR) return;
    float sg = sigma[0];
    float inv_s2 = 1.0f / (sg * sg);
    const float* r = mu + (size_t)n * DKEY;
    float m2 = 0.0f;
#pragma unroll
    for (int k = 0; k < DKEY; k += 4) {
        float4 v = *(const float4*)(r + k);
        m2 += v.x * v.x + v.y * v.y + v.z * v.z + v.w * v.w;
    }
    bias[n] = -0.5f * m2 * inv_s2;
}

// ---------------------------------------------------------------------------
// Kernel 1: fused scores-GEMM (f32 WMMA, async double-buffered LDS staging)
//           + running top-32 + softmax + V/E gather-reduce + gate.
//           One wave owns 16 query rows; block stages 64-center tiles.
// ---------------------------------------------------------------------------
__global__ __launch_bounds__(128)
void memattn_main_kernel(const float* __restrict__ x,
                         const float* __restrict__ q,
                         const float* __restrict__ gprior,
                         const float* __restrict__ mu,
                         const float* __restrict__ Vmem,
                         const float* __restrict__ Emem,
                         const float* __restrict__ sigma,
                         const float* __restrict__ Wg,
                         const float* __restrict__ Wgb,
                         const float* __restrict__ gpw,
                         const float* __restrict__ bias,
                         float* __restrict__ out_rV,
                         float* __restrict__ out_rE,
                         float* __restrict__ out_g) {
    __shared__ float ldsMu[2][TILE_N * MU_STRIDE];   // 2 x 17408 B
    __shared__ float ldsBias[2][TILE_N];
    __shared__ float listS[WAVES][16][TOPK];
    __shared__ int   listI[WAVES][16][TOPK];
    __shared__ float rowMin[WAVES][16];
    __shared__ float wbuf[WAVES][TOPK];
    __shared__ int   ibuf[WAVES][TOPK];

    const int tid   = threadIdx.x;
    const int wv    = tid >> 5;     // wave in block (0..3)
    const int lane  = tid & 31;
    const int half  = lane >> 4;    // f32 WMMA: half selects K pair / M+8 rows
    const int l16   = lane & 15;
    const int kbase = half * 2;
    const int blockRow = blockIdx.x * (WAVES * 16);

    const float sg = sigma[0];
    const float inv_s2 = 1.0f / (sg * sg);

    // init per-wave top-k state (wave-private LDS, in-order DS => no barrier)
#pragma unroll
    for (int row = 0; row < 16; ++row)
        listS[wv][row][lane] = -3.4e38f;
    if (lane < 16) rowMin[wv][lane] = -3.4e38f;

    // A-matrix (16 rows x 64 K, f32 WMMA layout) held in registers throughout
    v2f a[16];
    {
        const int aRow = blockRow + wv * 16 + l16;
        const float* qr = q + (size_t)aRow * DKEY + kbase;
#pragma unroll
        for (int kc = 0; kc < 16; ++kc)
            a[kc] = *(const v2f*)(qr + 4 * kc);
    }

    // stage helper: 9 async ops per thread per tile (8 x b128 mu + 1 x b32 bias)
    auto stage_async = [&](int tile, int buf) {
        const float* gsrc = mu + (size_t)tile * TILE_N * DKEY;
#pragma unroll
        for (int j = 0; j < 8; ++j) {
            int f = (j * 128 + tid) * 4;            // 0..4095 step 4
            async_g2l_b128(gsrc + f,
                           &ldsMu[buf][(f >> 6) * MU_STRIDE + (f & 63)]);
        }
        // all 128 threads issue; tids 64..127 duplicate-write identical data
        async_g2l_b32(bias + tile * TILE_N + (tid & 63),
                      &ldsBias[buf][tid & 63]);
    };

    // top-32 maintenance for one 16x16 score sub-tile
    auto topk_update = [&](const v8f& c, int cbase, float bcol) {
#pragma unroll
        for (int r = 0; r < 8; ++r) {
            const int   row = r + half * 8;         // C/D layout: halves M / M+8
            const float s   = c[r] * inv_s2 + bcol;
            bool want = s > rowMin[wv][row];
            unsigned long long m = __ballot(want);
            while (m) {
                int src = (int)__ffsll(m) - 1;
                if (lane == src) {
                    float mn = listS[wv][row][0];
                    int   mi = 0;
                    for (int e = 1; e < TOPK; ++e) {
                        float v = listS[wv][row][e];
                        if (v < mn) { mn = v; mi = e; }
                    }
                    if (s > mn) {
                        listS[wv][row][mi] = s;
                        listI[wv][row][mi] = cbase + l16;
                        float nm = 3.4e38f;
                        for (int e = 0; e < TOPK; ++e)
                            nm = fminf(nm, listS[wv][row][e]);
                        rowMin[wv][row] = nm;
                    }
                }
                m &= m - 1;
            }
        }
    };

    // prologue: kick off tile 0 into buffer 0
    stage_async(0, 0);

    for (int nb = 0; nb < NITER; ++nb) {
        const int cur = nb & 1;
        // issue next tile into the other buffer, then wait for current tile:
        // async loads complete in order, so asynccnt<=9 => current 9 are done.
        if (nb + 1 < NITER) {
            stage_async(nb + 1, cur ^ 1);
            asm volatile("s_wait_asynccnt 0x9" ::: "memory");
        } else {
            asm volatile("s_wait_asynccnt 0x0" ::: "memory");
        }
        __syncthreads();            // tile nb visible to all waves

        if (nb + 2 < NITER)         // warm L2 two tiles ahead
            __builtin_prefetch(mu + (size_t)(nb + 2) * TILE_N * DKEY + tid * 32, 0, 3);

        // 64 centers = 4 sub-tiles; two interleaved WMMA accumulator chains
#pragma unroll
        for (int p = 0; p < 2; ++p) {
            v8f c0 = {0.f,0.f,0.f,0.f,0.f,0.f,0.f,0.f};
            v8f c1 = {0.f,0.f,0.f,0.f,0.f,0.f,0.f,0.f};
            const float* b0p = &ldsMu[cur][(p * 32 + l16) * MU_STRIDE + kbase];
            const float* b1p = b0p + 16 * MU_STRIDE;
#pragma unroll
            for (int kc = 0; kc < 16; ++kc) {
                v2f b0 = *(const v2f*)(b0p + 4 * kc);
                v2f b1 = *(const v2f*)(b1p + 4 * kc);
                c0 = __builtin_amdgcn_wmma_f32_16x16x4_f32(false, a[kc], false, b0,
                                                           (short)0, c0, false, false);
                c1 = __builtin_amdgcn_wmma_f32_16x16x4_f32(false, a[kc], false, b1,
                                                           (short)0, c1, false, false);
            }
            topk_update(c0, nb * TILE_N + p * 32,
                        ldsBias[cur][p * 32 + l16]);
            topk_update(c1, nb * TILE_N + p * 32 + 16,
                        ldsBias[cur][p * 32 + 16 + l16]);
        }
        __syncthreads();            // all reads of buf[cur] done -> refill safe
    }

    // ------------------------- epilogue: one row per pass --------------------
    for (int row = 0; row < 16; ++row) {
        const int grow = blockRow + wv * 16 + row;
        float s  = listS[wv][row][lane];
        int   id = listI[wv][row][lane];

        // softmax over the 32 kept scores (wave-wide reductions)
        float mx = s;
#pragma unroll
        for (int o = 16; o > 0; o >>= 1) mx = fmaxf(mx, __shfl_xor(mx, o, 32));
        float wt = __expf(s - mx);
        float sum = wt;
#pragma unroll
        for (int o = 16; o > 0; o >>= 1) sum += __shfl_xor(sum, o, 32);
        wt /= sum;
        wbuf[wv][lane] = wt;
        ibuf[wv][lane] = id;        // same-wave LDS: in-order, no barrier needed

        // r_V: lane owns dims [4*lane, 4*lane+4); coalesced float4 gathers
        float4 acc = {0.f, 0.f, 0.f, 0.f};
        for (int k = 0; k < TOPK; ++k) {
            float wk = wbuf[wv][k];
            const float* vr = Vmem + (size_t)ibuf[wv][k] * DVAL + lane * 4;
            float4 v = *(const float4*)vr;
            acc.x += wk * v.x; acc.y += wk * v.y;
            acc.z += wk * v.z; acc.w += wk * v.w;
        }
        *(float4*)(out_rV + (size_t)grow * DVAL + lane * 4) = acc;

        // r_E: each lane weights its own entry, reduce 4-vector across wave
        float4 e = *(const float4*)(Emem + (size_t)id * 4);
        float ex = wt * e.x, ey = wt * e.y, ez = wt * e.z, ew = wt * e.w;
#pragma unroll
        for (int o = 16; o > 0; o >>= 1) {
            ex += __shfl_xor(ex, o, 32); ey += __shfl_xor(ey, o, 32);
            ez += __shfl_xor(ez, o, 32); ew += __shfl_xor(ew, o, 32);
        }
        if (lane == 0) {
            float* oe = out_rE + (size_t)grow * 4;
            oe[0] = ex; oe[1] = ey; oe[2] = ez; oe[3] = ew;
        }

        // gate: sigmoid( [x ; r_V] . Wg + b + gpw * g_prior )
        const float* xr = x + (size_t)grow * DMODEL;
        float part = 0.f;
#pragma unroll
        for (int j = 0; j < 8; ++j) {
            int d = lane + 32 * j;
            part += xr[d] * Wg[d];
        }
        const float* Wv = Wg + DMODEL + lane * 4;
        part += acc.x * Wv[0] + acc.y * Wv[1] + acc.z * Wv[2] + acc.w * Wv[3];
#pragma unroll
        for (int o = 16; o > 0; o >>= 1) part += __shfl_xor(part, o, 32);
        if (lane == 0) {
            float z = part + Wgb[0] + gpw[0] * gprior[grow];
            out_g[grow] = 1.0f / (1.0f + __expf(-z));
        }
    }
}

// ---------------------------------------------------------------------------
extern "C" void kernel_launch(void* const* d_in, const int* in_sizes, int n_in,
                              void* d_out, int out_size, void* d_ws, size_t ws_size,
                              hipStream_t stream) {
    const float* x      = (const float*)d_in[0];   // [B,T,256]
    const float* q      = (const float*)d_in[1];   // [B,T,64]
    const float* gprior = (const float*)d_in[2];   // [B,T,1]
    const float* mu     = (const float*)d_in[3];   // [N,64]
    const float* Vmem   = (const float*)d_in[4];   // [N,128]
    const float* Emem   = (const float*)d_in[5];   // [N,4]
    const float* sigma  = (const float*)d_in[6];   // scalar
    const float* Wg     = (const float*)d_in[7];   // [1,384]
    const float* Wgb    = (const float*)d_in[8];   // [1]
    const float* gpw    = (const float*)d_in[9];   // scalar
    (void)in_sizes; (void)n_in; (void)out_size; (void)ws_size;

    float* bias = (float*)d_ws;                    // 32768 floats = 128 KB

    float* out    = (float*)d_out;
    float* out_rV = out;                                          // 8192*128
    float* out_rE = out + (size_t)ROWS_TOTAL * DVAL;              // 8192*4
    float* out_g  = out_rE + (size_t)ROWS_TOTAL * 4;              // 8192

    memattn_bias_kernel<<<NCTR / 128, 128, 0, stream>>>(mu, sigma, bias);
    memattn_main_kernel<<<ROWS_TOTAL / (WAVES * 16), 128, 0, stream>>>(
        x, q, gprior, mu, Vmem, Emem, sigma, Wg, Wgb, gpw, bias,
        out_rV, out_rE, out_g);
}